// StargazerGNN_46694884442368
// MI455X (gfx1250) — compile-verified
//
#include <hip/hip_runtime.h>

#define N_NODES 100000
#define N_EDGES 1600000
#define N_GRAPHS 64
#define HID 128

typedef __attribute__((ext_vector_type(2))) float    v2f;
typedef __attribute__((ext_vector_type(8))) float    v8f;
typedef __attribute__((ext_vector_type(4))) unsigned v4u;
typedef __attribute__((ext_vector_type(4))) int      v4i;
typedef __attribute__((ext_vector_type(8))) int      v8i;

// ---------------------------------------------------------------- utilities
__global__ void zero_kernel(float* __restrict__ p, int count) {
    int i = blockIdx.x * blockDim.x + threadIdx.x;
    if (i < count) p[i] = 0.0f;
}

// ------------------------------------------------------------------ degrees
__global__ void deg_kernel(const int* __restrict__ src, const int* __restrict__ dst,
                           float* __restrict__ deg_in, float* __restrict__ deg_out, int E) {
    int e = blockIdx.x * blockDim.x + threadIdx.x;
    if (e < E) {
        atomicAdd(&deg_in[dst[e]], 1.0f);
        atomicAdd(&deg_out[src[e]], 1.0f);
    }
}

__global__ void norm_kernel(const float* __restrict__ deg_in, const float* __restrict__ deg_out,
                            float* __restrict__ norm_src, float* __restrict__ norm_dst, int n) {
    int i = blockIdx.x * blockDim.x + threadIdx.x;
    if (i < n) {
        norm_src[i] = rsqrtf(fmaxf(deg_out[i], 1.0f));
        norm_dst[i] = rsqrtf(fmaxf(deg_in[i], 1.0f));
    }
}

// ------------------------------------------------- layer 1 (feature dim 1)
__global__ void scatter1_kernel(const int* __restrict__ src, const int* __restrict__ dst,
                                const float* __restrict__ h0, const float* __restrict__ norm_src,
                                float* __restrict__ agg, int E) {
    int e = blockIdx.x * blockDim.x + threadIdx.x;
    if (e < E) {
        int s = src[e];
        atomicAdd(&agg[dst[e]], h0[s] * norm_src[s]);
    }
}

// h[n][j] = relu(agg[n]*norm_dst[n]*W1[j] + b1[j])   (W1 is 1x128)
__global__ void layer1_kernel(const float* __restrict__ agg, const float* __restrict__ norm_dst,
                              const float* __restrict__ W1, const float* __restrict__ b1,
                              float* __restrict__ hout, int n) {
    int idx = blockIdx.x * blockDim.x + threadIdx.x;
    if (idx < n * HID) {
        int node = idx >> 7, j = idx & 127;
        float x = agg[node] * norm_dst[node];
        float v = x * W1[j] + b1[j];
        hout[idx] = v > 0.0f ? v : 0.0f;
    }
}

// ------------------------------------- edge scatter-add for 128-wide layers
// One wave per edge; lane l moves cols [4l, 4l+3] (float4 read, 4 f32 atomics).
__global__ __launch_bounds__(256) void scatter128_kernel(
        const int* __restrict__ src, const int* __restrict__ dst,
        const float* __restrict__ norm_src, const float* __restrict__ h,
        float* __restrict__ agg, int E) {
    int gid  = blockIdx.x * blockDim.x + threadIdx.x;
    int edge = gid >> 5;
    int lane = gid & 31;
    if (edge >= E) return;
    int s = src[edge];
    int d = dst[edge];
    float ns = norm_src[s];
    float4 v = ((const float4*)(h + (size_t)s * HID))[lane];
    float* arow = agg + (size_t)d * HID + lane * 4;
    atomicAdd(arow + 0, v.x * ns);
    atomicAdd(arow + 1, v.y * ns);
    atomicAdd(arow + 2, v.z * ns);
    atomicAdd(arow + 3, v.w * ns);
}

// ------------------------------------------- WMMA GEMM + bias + ReLU layer
// hout = relu((agg * norm_dst[:,None]) @ W + b)
//      = relu(norm_dst[m] * (agg @ W)[m,n] + b[n])   (row scale folded to epilogue)
// Block = 256 threads = 8 waves; 16 rows x 128 cols per block; wave w owns cols
// [16w,16w+16). A-tile (16x128 f32, contiguous 8KB of agg) is DMA'd to LDS by
// the Tensor Data Mover with hardware padding (+4 dwords per 128) so fragment
// reads are bank-conflict-free. K-loop: 32 x V_WMMA_F32_16X16X4_F32.
__global__ __launch_bounds__(256) void gemm_relu_kernel(
        const float* __restrict__ agg, const float* __restrict__ norm_dst,
        const float* __restrict__ W, const float* __restrict__ b,
        float* __restrict__ hout) {
    __shared__ float As[16][HID + 4];   // matches TDM pad: 132 dwords per row

    const int row0 = blockIdx.x * 16;
    const int tid  = threadIdx.x;

    if (tid < 32) {  // wave 0 issues one TDM descriptor for the whole tile
        unsigned lds_off = (unsigned)(size_t)&As[0][0];           // addr[31:0] = LDS offset
        unsigned long long ga = (unsigned long long)(const void*)(agg + (size_t)row0 * HID);
        // D# group 0: count=1 | lds_addr | global_addr[56:0] | type=2
        v4u g0 = { 1u,
                   lds_off,
                   (unsigned)(ga & 0xFFFFFFFFu),
                   (unsigned)((ga >> 32) & 0x01FFFFFFu) | 0x80000000u };
        // D# group 1: data_size=4B(2), pad_enable, pad_interval=128dw(6),
        // pad_amount=4dw(3); tensor_dim0=tile_dim0=2048 dwords (1-D copy),
        // tensor_dim1=1, stride0=2048.
        v8i g1 = { (int)0x07920000,
                   (int)0x08000000,   // tensor_dim0[15:0]=2048 << 16
                   (int)0x00010000,   // tensor_dim1[15:0]=1   << 16
                   (int)0x08000000,   // tile_dim0=2048        << 16
                   0,                 // tile_dim1/2 unused
                   2048,              // tensor_dim0_stride
                   0, 0 };
        v4i gz = { 0, 0, 0, 0 };
#if defined(__clang_major__) && __clang_major__ >= 23
        v8i gz8 = { 0, 0, 0, 0, 0, 0, 0, 0 };
        __builtin_amdgcn_tensor_load_to_lds(g0, g1, gz, gz, gz8, 0);
#else
        __builtin_amdgcn_tensor_load_to_lds(g0, g1, gz, gz, 0);
#endif
        __builtin_amdgcn_s_wait_tensorcnt(0);
    }
    // W stays hot in the 192MB L2 (64KB total); hint the near levels.
    __builtin_prefetch(&W[(tid & 31) * 4], 0, 3);
    __syncthreads();

    const int wave  = tid >> 5;
    const int lane  = tid & 31;
    const int ntile = wave * 16;
    const int m     = lane & 15;          // A row / B,C col within tile
    const int kb    = (lane >> 4) * 2;    // K sub-offset per ISA v2f layout
    const int ncol  = ntile + m;

    v8f acc = {};
    #pragma unroll
    for (int ks = 0; ks < 32; ++ks) {
        const int k0 = ks * 4 + kb;
        v2f a, bb;
        a.x  = As[m][k0];
        a.y  = As[m][k0 + 1];
        bb.x = W[k0 * HID + ncol];
        bb.y = W[(k0 + 1) * HID + ncol];
        acc = __builtin_amdgcn_wmma_f32_16x16x4_f32(
            /*neg_a=*/false, a, /*neg_b=*/false, bb,
            /*c_mod=*/(short)0, acc, /*reuse_a=*/false, /*reuse_b=*/false);
    }

    // Epilogue: D VGPR r -> row (r + 8*(lane>>4)), col (lane&15) of the tile.
    // Apply the per-row symmetric norm here, then bias + ReLU.
    const float bias = b[ncol];
    const int mhi = (lane >> 4) * 8;
    #pragma unroll
    for (int r = 0; r < 8; ++r) {
        const int row = row0 + r + mhi;
        float v = acc[r] * norm_dst[row] + bias;
        v = v > 0.0f ? v : 0.0f;
        hout[row * HID + ncol] = v;
    }
}

// -------------------------------------------------------- graph mean pool
__global__ void pool_kernel(const int* __restrict__ graph_ids, const float* __restrict__ h,
                            float* __restrict__ pool, float* __restrict__ counts, int n) {
    int idx = blockIdx.x * blockDim.x + threadIdx.x;
    if (idx >= n * 32) return;
    int node = idx >> 5, q = idx & 31;
    int g = graph_ids[node];
    float4 v = ((const float4*)(h + (size_t)node * HID))[q];
    float* p = pool + g * HID + q * 4;
    atomicAdd(p + 0, v.x);
    atomicAdd(p + 1, v.y);
    atomicAdd(p + 2, v.z);
    atomicAdd(p + 3, v.w);
    if (q == 0) atomicAdd(&counts[g], 1.0f);
}

// ------------------------------------------------- classifier: 128 -> 2
__global__ void classify_kernel(const float* __restrict__ pool, const float* __restrict__ counts,
                                const float* __restrict__ Wc, const float* __restrict__ bc,
                                float* __restrict__ out) {
    int t = threadIdx.x;              // 128 threads: (g, c)
    if (t >= N_GRAPHS * 2) return;
    int g = t >> 1, c = t & 1;
    float inv = 1.0f / fmaxf(counts[g], 1.0f);
    float acc = bc[c];
    #pragma unroll 8
    for (int j = 0; j < HID; ++j)
        acc += pool[g * HID + j] * inv * Wc[j * 2 + c];
    out[g * 2 + c] = acc;
}

// --------------------------------------------------------------- launcher
extern "C" void kernel_launch(void* const* d_in, const int* in_sizes, int n_in,
                              void* d_out, int out_size, void* d_ws, size_t ws_size,
                              hipStream_t stream) {
    const int*   src       = (const int*)d_in[0];
    const int*   dst       = (const int*)d_in[1];
    const int*   graph_ids = (const int*)d_in[2];
    const float* W1 = (const float*)d_in[3];
    const float* b1 = (const float*)d_in[4];
    const float* W2 = (const float*)d_in[5];
    const float* b2 = (const float*)d_in[6];
    const float* W3 = (const float*)d_in[7];
    const float* b3 = (const float*)d_in[8];
    const float* Wc = (const float*)d_in[9];
    const float* bc = (const float*)d_in[10];
    float* out = (float*)d_out;

    const int N = N_NODES, E = N_EDGES, G = N_GRAPHS;

    // Workspace layout (floats). Total ~155 MB.
    float* ws       = (float*)d_ws;
    float* deg_in   = ws;                       // N   (also h0)
    float* deg_out  = deg_in  + N;              // N
    float* norm_src = deg_out + N;              // N
    float* norm_dst = norm_src + N;             // N
    float* hA       = norm_dst + N;             // N*128
    float* hB       = hA + (size_t)N * HID;     // N*128
    float* agg      = hB + (size_t)N * HID;     // N*128
    float* pool     = agg + (size_t)N * HID;    // G*128
    float* counts   = pool + G * HID;           // G

    const int T = 256;
    #define NB(x) (((x) + T - 1) / T)

    // Degrees + norms
    zero_kernel<<<NB(2 * N), T, 0, stream>>>(deg_in, 2 * N);
    deg_kernel<<<NB(E), T, 0, stream>>>(src, dst, deg_in, deg_out, E);
    norm_kernel<<<NB(N), T, 0, stream>>>(deg_in, deg_out, norm_src, norm_dst, N);

    // Layer 1 (dim 1 -> 128): h0 = deg_in
    zero_kernel<<<NB(N), T, 0, stream>>>(agg, N);
    scatter1_kernel<<<NB(E), T, 0, stream>>>(src, dst, deg_in, norm_src, agg, E);
    layer1_kernel<<<NB(N * HID), T, 0, stream>>>(agg, norm_dst, W1, b1, hA, N);

    // Layer 2 (128 -> 128): hA -> hB
    zero_kernel<<<NB(N * HID), T, 0, stream>>>(agg, N * HID);
    scatter128_kernel<<<NB(E * 32), T, 0, stream>>>(src, dst, norm_src, hA, agg, E);
    gemm_relu_kernel<<<N / 16, T, 0, stream>>>(agg, norm_dst, W2, b2, hB);

    // Layer 3 (128 -> 128): hB -> hA
    zero_kernel<<<NB(N * HID), T, 0, stream>>>(agg, N * HID);
    scatter128_kernel<<<NB(E * 32), T, 0, stream>>>(src, dst, norm_src, hB, agg, E);
    gemm_relu_kernel<<<N / 16, T, 0, stream>>>(agg, norm_dst, W3, b3, hA);

    // Mean pool + classifier
    zero_kernel<<<NB(G * HID + G), T, 0, stream>>>(pool, G * HID + G);
    pool_kernel<<<NB(N * 32), T, 0, stream>>>(graph_ids, hA, pool, counts, N);
    classify_kernel<<<1, 128, 0, stream>>>(pool, counts, Wc, bc, out);

    #undef NB
}